// HGA_41996190220397
// MI455X (gfx1250) — compile-verified
//
#include <hip/hip_runtime.h>

// ---------------------------------------------------------------------------
// HGA hetero-GNN pipeline for MI455X (gfx1250, wave32).
// Dense node-feature GEMMs run on V_WMMA_F32_16X16X4_F32 (native fp32 WMMA),
// 4 output tiles (16x64) per wave, software-pipelined so the next k-step's
// A/B fragments are in flight while the current 4-WMMA burst executes.
// Edge softmax/scatter passes are bandwidth-bound -> VALU + f32 atomics.
// ---------------------------------------------------------------------------

typedef float v2f __attribute__((ext_vector_type(2)));
typedef float v8f __attribute__((ext_vector_type(8)));

#define CDIV(a, b) (((a) + (b) - 1) / (b))

enum {
  N_PW = 50000, N_FG = 10000, BATCH = 512,
  E_PW = 400000, E_FG = 80000, NNZ = 200000,
  IN_F = 128, FG_F = 73, HDIM = 256, EDIMC = 11, CCH = 32
};

// order-preserving float <-> uint mapping (for atomicMax on floats)
__device__ __forceinline__ unsigned f2ord(float f) {
  unsigned u = __float_as_uint(f);
  return (u & 0x80000000u) ? ~u : (u | 0x80000000u);
}
__device__ __forceinline__ float ord2f(unsigned u) {
  return (u & 0x80000000u) ? __uint_as_float(u & 0x7fffffffu) : __uint_as_float(~u);
}
#define ORD_NEG_INF 0x007FFFFFu  // f2ord(-inf)

__device__ __forceinline__ float wave_sum(float v) {
  #pragma unroll
  for (int o = 16; o > 0; o >>= 1) v += __shfl_xor(v, o, 32);
  return v;
}

// ---------------------------------------------------------------------------
// GEMM: C[M,N] = (A (+A2)) [M,K] @ W [K,N] (+bias[N]).
// One wave computes a 16x(16*NT) tile; per k-step one A fragment feeds NT
// WMMAs. Loads for step k+4 are issued before the step-k WMMA burst.
// M mult of 16, N mult of 16*NT, K arbitrary (masked tail handles K=73).
// Fragment maps (per ISA 7.12.2, 32-bit operands):
//   A 16x4 : lane(0..15)=row, hi-half lanes hold K+2/K+3 in vgpr0/1
//   B 4x16 : lane(0..15)=col, vgpr g holds K row g (+2 for hi half)
//   C 16x16: vgpr g = row g + 8*hi, lane lo = col
// ---------------------------------------------------------------------------
template <int NT, bool HAS_A2>
__global__ void gemm16_wmma(const float* __restrict__ A, const float* __restrict__ A2,
                            const float* __restrict__ W, const float* __restrict__ bias,
                            float* __restrict__ C, int M, int K, int N) {
  int gw = (blockIdx.x * blockDim.x + threadIdx.x) >> 5;
  int lane = threadIdx.x & 31;
  int ng = N / (16 * NT);
  int tiles = (M >> 4) * ng;
  if (gw >= tiles) return;
  int mt = gw / ng, ntg = gw - mt * ng;
  int hi = lane >> 4, lo = lane & 15;
  long arow = (long)(mt * 16 + lo) * K;
  int col0 = ntg * (16 * NT) + lo;

  v8f acc[NT];
  #pragma unroll
  for (int t = 0; t < NT; ++t) acc[t] = {};

  // gather one k-step's fragments (A 16x4 + NT B 4x16 tiles) into registers
  auto loadk = [&](int kk, v2f& aa, v2f (&bb)[NT]) {
    int ka = kk + 2 * hi;
    aa.x = A[arow + ka];
    aa.y = A[arow + ka + 1];
    if (HAS_A2) {
      aa.x += A2[arow + ka];
      aa.y += A2[arow + ka + 1];
    }
    const float* wr0 = W + (long)ka * N + col0;
    const float* wr1 = wr0 + N;
    #pragma unroll
    for (int t = 0; t < NT; ++t) {
      bb[t].x = wr0[16 * t];
      bb[t].y = wr1[16 * t];
    }
  };

  int kmain = K & ~3;
  v2f a, b[NT];
  if (kmain) {
    loadk(0, a, b);  // prologue
    int k = 0;
    for (; k + 4 < kmain; k += 4) {
      v2f an, bn[NT];
      loadk(k + 4, an, bn);  // next step's loads go in flight first
      #pragma unroll
      for (int t = 0; t < NT; ++t)
        acc[t] = __builtin_amdgcn_wmma_f32_16x16x4_f32(false, a, false, b[t], (short)0,
                                                       acc[t], false, false);
      a = an;
      #pragma unroll
      for (int t = 0; t < NT; ++t) b[t] = bn[t];
    }
    #pragma unroll
    for (int t = 0; t < NT; ++t)  // epilogue burst
      acc[t] = __builtin_amdgcn_wmma_f32_16x16x4_f32(false, a, false, b[t], (short)0,
                                                     acc[t], false, false);
  }
  if (kmain < K) {  // masked tail (K=73 case)
    int ka = kmain + 2 * hi;
    v2f at = {0.f, 0.f};
    bool m0 = ka < K, m1 = ka + 1 < K;
    if (m0) at.x = A[arow + ka] + (HAS_A2 ? A2[arow + ka] : 0.f);
    if (m1) at.y = A[arow + ka + 1] + (HAS_A2 ? A2[arow + ka + 1] : 0.f);
    #pragma unroll
    for (int t = 0; t < NT; ++t) {
      v2f bt = {0.f, 0.f};
      if (m0) bt.x = W[(long)ka * N + col0 + 16 * t];
      if (m1) bt.y = W[(long)(ka + 1) * N + col0 + 16 * t];
      acc[t] = __builtin_amdgcn_wmma_f32_16x16x4_f32(false, at, false, bt, (short)0,
                                                     acc[t], false, false);
    }
  }

  long cbase = (long)(mt * 16) * N + col0;
  #pragma unroll
  for (int t = 0; t < NT; ++t) {
    float bv = bias ? bias[col0 + 16 * t] : 0.f;
    #pragma unroll
    for (int g = 0; g < 8; ++g)
      C[cbase + (long)(g + 8 * hi) * N + 16 * t] = acc[t][g] + bv;
  }
}

static void launch_gemm(hipStream_t st, const float* A, const float* A2,
                        const float* W, const float* bias, float* C,
                        int M, int K, int N) {
  int tiles = (M / 16) * (N / 64);  // NT=4 -> 16x64 per wave; N is 256 or 128
  int thr = tiles * 32;
  if (A2)
    gemm16_wmma<4, true><<<CDIV(thr, 256), 256, 0, st>>>(A, A2, W, bias, C, M, K, N);
  else
    gemm16_wmma<4, false><<<CDIV(thr, 256), 256, 0, st>>>(A, nullptr, W, bias, C, M, K, N);
}

// -------------------------- utility kernels --------------------------------
__global__ void fill_f32(float* p, float v, long n) {
  long i = (long)blockIdx.x * blockDim.x + threadIdx.x;
  if (i < n) p[i] = v;
}
__global__ void fill_u32(unsigned* p, unsigned v, int n) {
  int i = blockIdx.x * blockDim.x + threadIdx.x;
  if (i < n) p[i] = v;
}
__global__ void bias_fill(float* out, const float* __restrict__ b, int N, int H) {
  long i = (long)blockIdx.x * blockDim.x + threadIdx.x;
  if (i < (long)N * H) out[i] = b[i % H];
}
__global__ void lrelu_ker(float* p, long n, float slope) {
  long i = (long)blockIdx.x * blockDim.x + threadIdx.x;
  if (i >= n) return;
  float v = p[i];
  p[i] = v > 0.f ? v : slope * v;
}

// per-node dots: o1[n]=h[n]·a1, o2[n]=h[n]·a2 (a2/o2 optional). one wave/node.
__global__ void node_dots(const float* __restrict__ h, const float* __restrict__ a1,
                          const float* __restrict__ a2, float* o1, float* o2,
                          int N, int H) {
  int w = (blockIdx.x * blockDim.x + threadIdx.x) >> 5;
  int lane = threadIdx.x & 31;
  if (w >= N) return;
  float s1 = 0.f, s2 = 0.f;
  for (int i = lane; i < H; i += 32) {
    float v = h[(long)w * H + i];
    s1 += v * a1[i];
    if (a2) s2 += v * a2[i];
  }
  s1 = wave_sum(s1);
  if (a2) s2 = wave_sum(s2);
  if (!lane) {
    o1[w] = s1;
    if (o2) o2[w] = s2;
  }
}

// ve[r] = sum_h We[r][h]*ae[h]  (collapses (ea@We)@a_e to ea@ve). wave/row.
__global__ void matvecT(const float* __restrict__ We, const float* __restrict__ ae,
                        float* ve, int R, int H) {
  int r = blockIdx.x;
  if (r >= R) return;
  float s = 0.f;
  for (int i = threadIdx.x; i < H; i += 32) s += We[(long)r * H + i] * ae[i];
  s = wave_sum(s);
  if (!threadIdx.x) ve[r] = s;
}

__global__ void edge_attr_dot(const float* __restrict__ ea, const float* __restrict__ ve,
                              float* eadd, int E, int D) {
  int e = blockIdx.x * blockDim.x + threadIdx.x;
  if (e >= E) return;
  float s = 0.f;
  for (int i = 0; i < D; ++i) s += ea[(long)e * D + i] * ve[i];
  eadd[e] = s;
}

// logit = leaky_relu_{0.2}(aA[ia]+aB[ib]+eadd); segment-max into mbits[tgt]
__global__ void edge_logit_max(const int* __restrict__ ia, const int* __restrict__ ib,
                               const int* __restrict__ tgt, const float* __restrict__ aA,
                               const float* __restrict__ aB, const float* __restrict__ eadd,
                               float* elog, unsigned* mbits, int E) {
  int e = blockIdx.x * blockDim.x + threadIdx.x;
  if (e >= E) return;
  float v = aA[ia[e]] + aB[ib[e]] + (eadd ? eadd[e] : 0.f);
  v = v > 0.f ? v : 0.2f * v;
  elog[e] = v;
  atomicMax(&mbits[tgt[e]], f2ord(v));
}

__global__ void finalize_max(const unsigned* __restrict__ mbits, float* mval, int N) {
  int n = blockIdx.x * blockDim.x + threadIdx.x;
  if (n >= N) return;
  float f = ord2f(mbits[n]);
  mval[n] = isfinite(f) ? f : 0.f;  // empty segments -> 0 (matches reference)
}

__global__ void edge_exp_den(const int* __restrict__ tgt, const float* __restrict__ mval,
                             float* elog, float* den, int E) {
  int e = blockIdx.x * blockDim.x + threadIdx.x;
  if (e >= E) return;
  float ex = __expf(elog[e] - mval[tgt[e]]);
  elog[e] = ex;
  atomicAdd(&den[tgt[e]], ex);
}

// out[tgt] += alpha * msg[midx]; one wave per edge, lanes stride dims (coalesced)
__global__ void edge_scatter(const int* __restrict__ tgt, const int* __restrict__ midx,
                             const float* __restrict__ elog, const float* __restrict__ den,
                             const float* __restrict__ msg, float* out, int E, int H) {
  int w = (blockIdx.x * blockDim.x + threadIdx.x) >> 5;
  int lane = threadIdx.x & 31;
  if (w >= E) return;
  int t = tgt[w], m = midx[w];
  float al = elog[w] / fmaxf(den[t], 1e-16f);
  for (int i = lane; i < H; i += 32)
    atomicAdd(&out[(long)t * H + i], al * msg[(long)m * H + i]);
}

// -------------------------- pooling ----------------------------------------
__global__ void pool_scatter(const float* __restrict__ x, const int* __restrict__ batch,
                             float* s, unsigned* mxb, float* cnt, int N, int H) {
  int w = (blockIdx.x * blockDim.x + threadIdx.x) >> 5;
  int lane = threadIdx.x & 31;
  if (w >= N) return;
  int b = batch[w];
  for (int i = lane; i < H; i += 32) {
    float v = x[(long)w * H + i];
    atomicAdd(&s[(long)b * H + i], v);
    atomicMax(&mxb[(long)b * H + i], f2ord(v));
  }
  if (!lane) atomicAdd(&cnt[b], 1.f);
}

// g[b][3][16][16]: ch0=sum, ch1=mean, ch2=max (0 if empty)
__global__ void pool_assemble(const float* __restrict__ s, const unsigned* __restrict__ mxb,
                              const float* __restrict__ cnt, float* g, int B, int H) {
  int idx = blockIdx.x * blockDim.x + threadIdx.x;
  if (idx >= B * H) return;
  int b = idx / H, d = idx - b * H;
  float sv = s[idx], c = cnt[b];
  float mx = ord2f(mxb[idx]);
  g[((long)b * 3 + 0) * H + d] = sv;
  g[((long)b * 3 + 1) * H + d] = sv / fmaxf(c, 1.f);
  g[((long)b * 3 + 2) * H + d] = (c > 0.f) ? mx : 0.f;
}

// -------------------------- res block (16x16 images) ------------------------
__global__ void conv3x3(const float* __restrict__ in, const float* __restrict__ k,
                        float* out, int N, int Ci, int Co) {
  int idx = blockIdx.x * blockDim.x + threadIdx.x;
  if (idx >= N * Co * 256) return;
  int x = idx & 15, y = (idx >> 4) & 15;
  int co = (idx >> 8) % Co, n = (idx >> 8) / Co;
  float s = 0.f;
  for (int ci = 0; ci < Ci; ++ci)
    for (int dy = -1; dy <= 1; ++dy) {
      int yy = y + dy;
      if (yy < 0 || yy > 15) continue;
      for (int dx = -1; dx <= 1; ++dx) {
        int xx = x + dx;
        if (xx < 0 || xx > 15) continue;
        s += in[(((long)n * Ci + ci) * 16 + yy) * 16 + xx] *
             k[(((long)co * Ci + ci) * 3 + (dy + 1)) * 3 + (dx + 1)];
      }
    }
  out[idx] = s;
}

__global__ void bn2d_stats(const float* __restrict__ x, float* mu, float* rstd,
                           int N, int C) {
  int c = blockIdx.x;
  int cnt = N << 8;
  float s = 0.f, s2 = 0.f;
  for (int i = threadIdx.x; i < cnt; i += blockDim.x) {
    float v = x[((long)((i >> 8) * C + c) << 8) + (i & 255)];
    s += v; s2 += v * v;
  }
  __shared__ float sh[512];
  sh[threadIdx.x] = s; sh[256 + threadIdx.x] = s2;
  __syncthreads();
  for (int st = 128; st > 0; st >>= 1) {
    if ((int)threadIdx.x < st) {
      sh[threadIdx.x] += sh[threadIdx.x + st];
      sh[256 + threadIdx.x] += sh[256 + threadIdx.x + st];
    }
    __syncthreads();
  }
  if (!threadIdx.x) {
    float m = sh[0] / cnt;
    float var = sh[256] / cnt - m * m;
    mu[c] = m;
    rstd[c] = rsqrtf(var + 1e-5f);
  }
}

__global__ void bn2d_apply(float* x, const float* __restrict__ mu,
                           const float* __restrict__ rstd, long n, int C, int relu) {
  long i = (long)blockIdx.x * blockDim.x + threadIdx.x;
  if (i >= n) return;
  int c = (int)((i >> 8) % C);
  float v = (x[i] - mu[c]) * rstd[c];
  x[i] = relu ? fmaxf(v, 0.f) : v;
}

// yacc (+)= relu(t2 + 1x1skip(xin))   xin: [N,3,16,16], ks: [Co,3]
__global__ void res_skip(const float* __restrict__ t2, const float* __restrict__ xin,
                         const float* __restrict__ ks, float* yacc, int N, int Co, int acc) {
  int idx = blockIdx.x * blockDim.x + threadIdx.x;
  if (idx >= N * Co * 256) return;
  int p = idx & 255, co = (idx >> 8) % Co, n = (idx >> 8) / Co;
  float sk = 0.f;
  #pragma unroll
  for (int ci = 0; ci < 3; ++ci)
    sk += xin[(((long)n * 3 + ci) << 8) + p] * ks[co * 3 + ci];
  float v = fmaxf(t2[idx] + sk, 0.f);
  if (acc) yacc[idx] += v; else yacc[idx] = v;
}

__global__ void chan_mean(const float* __restrict__ y, float* feat, int N, int Co) {
  int idx = blockIdx.x * blockDim.x + threadIdx.x;
  if (idx >= N * 256) return;
  int n = idx >> 8, p = idx & 255;
  float s = 0.f;
  for (int co = 0; co < Co; ++co) s += y[(((long)n * Co + co) << 8) + p];
  feat[idx] = s / (float)Co;
}

// -------------------------- head -------------------------------------------
__global__ void bn1d_relu(float* t, int rows, int cols) {
  int j = blockIdx.x;
  float s = 0.f, s2 = 0.f;
  for (int r = threadIdx.x; r < rows; r += blockDim.x) {
    float v = t[(long)r * cols + j];
    s += v; s2 += v * v;
  }
  __shared__ float sh[512];
  __shared__ float mean, rstd;
  sh[threadIdx.x] = s; sh[256 + threadIdx.x] = s2;
  __syncthreads();
  for (int st = 128; st > 0; st >>= 1) {
    if ((int)threadIdx.x < st) {
      sh[threadIdx.x] += sh[threadIdx.x + st];
      sh[256 + threadIdx.x] += sh[256 + threadIdx.x + st];
    }
    __syncthreads();
  }
  if (!threadIdx.x) {
    mean = sh[0] / rows;
    float var = sh[256] / rows - mean * mean;
    rstd = rsqrtf(var + 1e-5f);
  }
  __syncthreads();
  for (int r = threadIdx.x; r < rows; r += blockDim.x) {
    float v = (t[(long)r * cols + j] - mean) * rstd;
    t[(long)r * cols + j] = fmaxf(v, 0.f);
  }
}

__global__ void head_out(const float* __restrict__ t, const float* __restrict__ W2,
                         const float* __restrict__ b2, float* out, int rows, int cols) {
  int w = (blockIdx.x * blockDim.x + threadIdx.x) >> 5;
  int lane = threadIdx.x & 31;
  if (w >= rows) return;
  float s = 0.f;
  for (int j = lane; j < cols; j += 32) s += t[(long)w * cols + j] * W2[j];
  s = wave_sum(s);
  if (!lane) out[w] = s + b2[0];
}

// ---------------------------------------------------------------------------
// Host-side layer drivers
// ---------------------------------------------------------------------------
struct Scr {
  float *sc1, *sc2, *mval, *den, *elog, *eadd, *ve;
  unsigned* mbits;
};

static void run_gat(hipStream_t st, const float* X, const float* Xadd, int N, int K,
                    const int* src, const int* dst, int E,
                    const float* W, const float* a_src, const float* a_dst,
                    const float* bvec, const float* We, const float* a_e,
                    const float* eattr, float* h, float* out, const Scr& s) {
  launch_gemm(st, X, Xadd, W, nullptr, h, N, K, HDIM);
  node_dots<<<CDIV(N * 32, 256), 256, 0, st>>>(h, a_src, a_dst, s.sc1, s.sc2, N, HDIM);
  const float* eaddp = nullptr;
  if (We) {
    matvecT<<<EDIMC, 32, 0, st>>>(We, a_e, s.ve, EDIMC, HDIM);
    edge_attr_dot<<<CDIV(E, 256), 256, 0, st>>>(eattr, s.ve, s.eadd, E, EDIMC);
    eaddp = s.eadd;
  }
  fill_u32<<<CDIV(N, 256), 256, 0, st>>>(s.mbits, ORD_NEG_INF, N);
  fill_f32<<<CDIV(N, 256), 256, 0, st>>>(s.den, 0.f, (long)N);
  edge_logit_max<<<CDIV(E, 256), 256, 0, st>>>(src, dst, dst, s.sc1, s.sc2, eaddp,
                                               s.elog, s.mbits, E);
  finalize_max<<<CDIV(N, 256), 256, 0, st>>>(s.mbits, s.mval, N);
  edge_exp_den<<<CDIV(E, 256), 256, 0, st>>>(dst, s.mval, s.elog, s.den, E);
  bias_fill<<<CDIV(N * HDIM, 256), 256, 0, st>>>(out, bvec, N, HDIM);
  edge_scatter<<<CDIV(E * 32, 256), 256, 0, st>>>(dst, src, s.elog, s.den, h, out, E, HDIM);
  lrelu_ker<<<CDIV((long)N * HDIM, 256), 256, 0, st>>>(out, (long)N * HDIM, 0.01f);
}

static void run_hyg(hipStream_t st, const float* Xpw, const float* Xfg,
                    const int* row, const int* col,
                    const float* W, const float* We, const float* a_e, const float* a_n,
                    const float* bvec, bool atom, float* hn, float* he, float* out,
                    const Scr& s) {
  launch_gemm(st, Xpw, nullptr, W, nullptr, hn, N_PW, HDIM, HDIM);
  launch_gemm(st, Xfg, nullptr, We, nullptr, he, N_FG, HDIM, HDIM);
  node_dots<<<CDIV(N_PW * 32, 256), 256, 0, st>>>(hn, a_n, nullptr, s.sc1, nullptr,
                                                  N_PW, HDIM);
  node_dots<<<CDIV(N_FG * 32, 256), 256, 0, st>>>(he, a_e, nullptr, s.sc2, nullptr,
                                                  N_FG, HDIM);
  const int* tgt = atom ? col : row;
  const int* midx = atom ? row : col;
  const float* msg = atom ? hn : he;
  int num = atom ? N_FG : N_PW;
  fill_u32<<<CDIV(num, 256), 256, 0, st>>>(s.mbits, ORD_NEG_INF, num);
  fill_f32<<<CDIV(num, 256), 256, 0, st>>>(s.den, 0.f, (long)num);
  edge_logit_max<<<CDIV(NNZ, 256), 256, 0, st>>>(row, col, tgt, s.sc1, s.sc2, nullptr,
                                                 s.elog, s.mbits, NNZ);
  finalize_max<<<CDIV(num, 256), 256, 0, st>>>(s.mbits, s.mval, num);
  edge_exp_den<<<CDIV(NNZ, 256), 256, 0, st>>>(tgt, s.mval, s.elog, s.den, NNZ);
  bias_fill<<<CDIV(num * HDIM, 256), 256, 0, st>>>(out, bvec, num, HDIM);
  edge_scatter<<<CDIV(NNZ * 32, 256), 256, 0, st>>>(tgt, midx, s.elog, s.den, msg, out,
                                                    NNZ, HDIM);
  lrelu_ker<<<CDIV((long)num * HDIM, 256), 256, 0, st>>>(out, (long)num * HDIM, 0.01f);
}

static void run_pool(hipStream_t st, const float* X, const int* batch, int N,
                     float* spool, unsigned* mxpool, float* cntb, float* g) {
  fill_f32<<<CDIV(BATCH * HDIM, 256), 256, 0, st>>>(spool, 0.f, (long)BATCH * HDIM);
  fill_u32<<<CDIV(BATCH * HDIM, 256), 256, 0, st>>>(mxpool, ORD_NEG_INF, BATCH * HDIM);
  fill_f32<<<CDIV(BATCH, 256), 256, 0, st>>>(cntb, 0.f, (long)BATCH);
  pool_scatter<<<CDIV(N * 32, 256), 256, 0, st>>>(X, batch, spool, mxpool, cntb, N, HDIM);
  pool_assemble<<<CDIV(BATCH * HDIM, 256), 256, 0, st>>>(spool, mxpool, cntb, g,
                                                         BATCH, HDIM);
}

static void run_res(hipStream_t st, const float* g, const float* k1, const float* k2,
                    const float* ks, float* t1, float* t2, float* yacc,
                    float* mu, float* rstd, int acc) {
  int nel = BATCH * CCH * 256;
  conv3x3<<<CDIV(nel, 256), 256, 0, st>>>(g, k1, t1, BATCH, 3, CCH);
  bn2d_stats<<<CCH, 256, 0, st>>>(t1, mu, rstd, BATCH, CCH);
  bn2d_apply<<<CDIV(nel, 256), 256, 0, st>>>(t1, mu, rstd, (long)nel, CCH, 1);
  conv3x3<<<CDIV(nel, 256), 256, 0, st>>>(t1, k2, t2, BATCH, CCH, CCH);
  bn2d_stats<<<CCH, 256, 0, st>>>(t2, mu, rstd, BATCH, CCH);
  bn2d_apply<<<CDIV(nel, 256), 256, 0, st>>>(t2, mu, rstd, (long)nel, CCH, 0);
  res_skip<<<CDIV(nel, 256), 256, 0, st>>>(t2, g, ks, yacc, BATCH, CCH, acc);
}

// ---------------------------------------------------------------------------
extern "C" void kernel_launch(void* const* d_in, const int* in_sizes, int n_in,
                              void* d_out, int out_size, void* d_ws, size_t ws_size,
                              hipStream_t stream) {
  (void)in_sizes; (void)n_in; (void)out_size; (void)ws_size;
  const float* x_pw  = (const float*)d_in[0];
  const float* x_fg  = (const float*)d_in[1];
  const float* eattr = (const float*)d_in[2];
  const int* ei_pw   = (const int*)d_in[3];
  const int* ei_fg   = (const int*)d_in[4];
  const int* ei_hy   = (const int*)d_in[5];
  const int* bat_pw  = (const int*)d_in[6];
  const int* bat_fg  = (const int*)d_in[7];

  // params flattened as JAX tree leaves (dict keys sorted alphabetically)
  const float* P[57];
  for (int i = 0; i < 57; ++i) P[i] = (const float*)d_in[8 + i];
  // gat1..3 (edge=True): W,We,a_dst,a_e,a_src,b  at 6*i
  // gat4..6            : W,a_dst,a_src,b         at 18+4*i
  // hyg1..4            : W,We,a_e,a_n,b          at 30+5*i
  // pred: W1=50,W2=51,b1=52,b2=53 ; res: k1=54,k2=55,ks=56
  #define GATE(i) P[6*(i)+0], P[6*(i)+4], P[6*(i)+2], P[6*(i)+5], P[6*(i)+1], P[6*(i)+3]
  #define GATN(i) P[18+4*(i)+0], P[18+4*(i)+2], P[18+4*(i)+1], P[18+4*(i)+3]
  #define HYG(i)  P[30+5*(i)+0], P[30+5*(i)+1], P[30+5*(i)+2], P[30+5*(i)+3], P[30+5*(i)+4]

  const int *src_pw = ei_pw, *dst_pw = ei_pw + E_PW;
  const int *src_fg = ei_fg, *dst_fg = ei_fg + E_FG;
  const int *hrow = ei_hy, *hcol = ei_hy + NNZ;

  // ---- workspace bump allocator (256B aligned) ----
  char* wbase = (char*)d_ws;
  size_t off = 0;
  auto ALLOCF = [&](size_t nf) -> float* {
    float* p = (float*)(wbase + off);
    off += ((nf * sizeof(float) + 255) & ~(size_t)255);
    return p;
  };
  float* A   = ALLOCF((size_t)N_PW * HDIM);  // current pw features
  float* Bf  = ALLOCF((size_t)N_FG * HDIM);  // current fg features
  float* Cx  = ALLOCF((size_t)N_PW * HDIM);  // pw aux (x_pw_1 / x_pw_2)
  float* Dx  = ALLOCF((size_t)N_FG * HDIM);  // fg aux (x_fg_1 / x_fg_2)
  float* Eb  = ALLOCF((size_t)N_PW * HDIM);  // h scratch (pw)
  float* Fb  = ALLOCF((size_t)N_FG * HDIM);  // h scratch (fg)
  Scr s;
  s.sc1   = ALLOCF(N_PW);
  s.sc2   = ALLOCF(N_PW);
  s.mval  = ALLOCF(N_PW);
  s.den   = ALLOCF(N_PW);
  s.elog  = ALLOCF(E_PW);
  s.eadd  = ALLOCF(E_PW);
  s.ve    = ALLOCF(HDIM);
  s.mbits = (unsigned*)ALLOCF(N_PW);
  float* spool   = ALLOCF((size_t)BATCH * HDIM);
  unsigned* mxp  = (unsigned*)ALLOCF((size_t)BATCH * HDIM);
  float* cntb    = ALLOCF(BATCH);
  float* gpw     = ALLOCF((size_t)BATCH * 3 * HDIM);
  float* gfg     = ALLOCF((size_t)BATCH * 3 * HDIM);
  float* mu      = ALLOCF(CCH);
  float* rstd    = ALLOCF(CCH);
  // res/head buffers overlay the (dead by then) Eb/Fb GEMM scratch:
  float* t1   = Eb;                                  // 512*32*256
  float* t2   = Eb + (size_t)BATCH * CCH * 256;      // 512*32*256
  float* yacc = Eb + (size_t)2 * BATCH * CCH * 256;  // 512*32*256 (Eb holds 12.8M)
  float* feat = Fb;                                  // 512*256
  float* tprd = Fb + (size_t)BATCH * HDIM;           // 512*128

  // ---- graph stages ----
  run_gat(stream, x_pw, nullptr, N_PW, IN_F, src_pw, dst_pw, E_PW, GATE(0), eattr,
          Eb, A, s);                                            // gat1 -> A
  run_gat(stream, x_fg, nullptr, N_FG, FG_F, src_fg, dst_fg, E_FG, GATN(0),
          nullptr, nullptr, nullptr, Fb, Bf, s);                // gat4 -> Bf
  run_hyg(stream, A, Bf, hrow, hcol, HYG(0), true,  Eb, Fb, Dx, s);  // hyg1 -> x_fg_1
  run_hyg(stream, A, Bf, hrow, hcol, HYG(1), false, Eb, Fb, Cx, s);  // hyg2 -> x_pw_1
  run_gat(stream, A, Cx, N_PW, HDIM, src_pw, dst_pw, E_PW, GATE(1), eattr,
          Eb, A, s);                                            // gat2 -> A
  run_gat(stream, Bf, Dx, N_FG, HDIM, src_fg, dst_fg, E_FG, GATN(1),
          nullptr, nullptr, nullptr, Fb, Bf, s);                // gat5 -> Bf
  run_hyg(stream, A, Bf, hrow, hcol, HYG(2), true,  Eb, Fb, Dx, s);  // hyg3 -> x_fg_2
  run_hyg(stream, A, Bf, hrow, hcol, HYG(3), false, Eb, Fb, Cx, s);  // hyg4 -> x_pw_2
  run_gat(stream, A, Cx, N_PW, HDIM, src_pw, dst_pw, E_PW, GATE(2), eattr,
          Eb, A, s);                                            // gat3 -> A
  run_gat(stream, Bf, Dx, N_FG, HDIM, src_fg, dst_fg, E_FG, GATN(2),
          nullptr, nullptr, nullptr, Fb, Bf, s);                // gat6 -> Bf

  // ---- pooling ----
  run_pool(stream, A,  bat_pw, N_PW, spool, mxp, cntb, gpw);
  run_pool(stream, Bf, bat_fg, N_FG, spool, mxp, cntb, gfg);

  // ---- res blocks (shared params), summed ----
  run_res(stream, gpw, P[54], P[55], P[56], t1, t2, yacc, mu, rstd, 0);
  run_res(stream, gfg, P[54], P[55], P[56], t1, t2, yacc, mu, rstd, 1);
  chan_mean<<<CDIV(BATCH * 256, 256), 256, 0, stream>>>(yacc, feat, BATCH, CCH);

  // ---- prediction head ----
  launch_gemm(stream, feat, nullptr, P[50], P[52], tprd, BATCH, HDIM, 128);  // feat@W1+b1
  bn1d_relu<<<128, 256, 0, stream>>>(tprd, BATCH, 128);
  head_out<<<CDIV(BATCH * 32, 256), 256, 0, stream>>>(tprd, P[51], P[53],
                                                      (float*)d_out, BATCH, 128);
}